// DirGATConv_61942018343497
// MI455X (gfx1250) — compile-verified
//
#include <hip/hip_runtime.h>
#include <hip/hip_bf16.h>

// DirGATConv for MI455X (gfx1250, wave32).
// Pipeline:
//   1) gemm_wmma   : h1 = x@W1, h2 = x@W2 via V_WMMA_F32_16X16X4_F32 (fp32 WMMA)
//   2) scores      : es/ed per (node, head) for both convs
//   3) init_out    : d_out = (1-A)*b1 + A*b2 broadcast; m=-inf; denom=0
//   4) edge_max    : segment max of leaky_relu(es[s]+ed[d]) per (dst, head)   [x2]
//   5) edge_sum    : segment sum of exp(v - m[d])                              [x2]
//   6) edge_msg    : d_out[d] += scale * att * h[s]   (atomic scatter)         [x2]

typedef __attribute__((ext_vector_type(2))) float v2f;
typedef __attribute__((ext_vector_type(8))) float v8f;

#define F_IN   128
#define HEADS  4
#define C_OUT  32
#define F_OUT  128   // HEADS * C_OUT
#define ALPHA_ 0.5f
#define SLOPE_ 0.2f

__device__ __forceinline__ void atomicMaxFloat(float* addr, float val) {
    // Ordering trick: non-negative floats compare correctly as signed ints,
    // negative floats compare inversely as unsigned ints.
    if (val >= 0.0f) {
        atomicMax((int*)addr, __float_as_int(val));
    } else {
        atomicMin((unsigned int*)addr, (unsigned int)__float_as_int(val));
    }
}

__device__ __forceinline__ float lrelu(float v) {
    return v > 0.0f ? v : v * SLOPE_;
}

// ---------------------------------------------------------------------------
// 1) fp32 WMMA GEMM: one wave per 16x16 output tile.  K = 128 = 32 steps of 4.
//    A (16x4 f32) layout: lane&15 = M row; lane>>4 selects K pair {0,1}/{2,3}.
//    B (4x16 f32) layout: lane&15 = N col; lane>>4 selects K pair.
//    C/D: VGPR r -> (M = r + 8*(lane>>4), N = lane&15).
// ---------------------------------------------------------------------------
__global__ __launch_bounds__(32) void gemm_wmma_kernel(
    const float* __restrict__ X,
    const float* __restrict__ W1, const float* __restrict__ W2,
    float* __restrict__ H1, float* __restrict__ H2, int n)
{
    int tile = blockIdx.x;
    int conv = tile & 1;         tile >>= 1;
    int ntile = tile & 7;        // 8 column tiles of 16 (F_OUT = 128)
    int mtile = tile >> 3;

    const float* __restrict__ Wm = conv ? W2 : W1;
    float* __restrict__ Hm       = conv ? H2 : H1;

    int lane = threadIdx.x;      // wave32
    int half = lane >> 4;        // 0 or 1
    int l    = lane & 15;

    int rowA = mtile * 16 + l;
    if (rowA >= n) rowA = n - 1;           // clamp (no divergence before WMMA)
    int colB = ntile * 16 + l;

    v8f acc = {};
    for (int k = 0; k < F_IN; k += 4) {
        int ka = k + half * 2;
        v2f a, b;
        a.x = X[rowA * F_IN + ka];
        a.y = X[rowA * F_IN + ka + 1];
        b.x = Wm[ka * F_OUT + colB];
        b.y = Wm[(ka + 1) * F_OUT + colB];
        acc = __builtin_amdgcn_wmma_f32_16x16x4_f32(
            /*neg_a=*/false, a, /*neg_b=*/false, b,
            /*c_mod=*/(short)0, acc, /*reuse_a=*/false, /*reuse_b=*/false);
    }

    int rowBase = mtile * 16 + half * 8;
    int colOut  = ntile * 16 + l;
    #pragma unroll
    for (int r = 0; r < 8; ++r) {
        int row = rowBase + r;
        if (row < n) Hm[row * F_OUT + colOut] = acc[r];
    }
}

// ---------------------------------------------------------------------------
// 2) per-(node, head) attention logits for both convs
// ---------------------------------------------------------------------------
__global__ void scores_kernel(
    const float* __restrict__ H1, const float* __restrict__ H2,
    const float* __restrict__ as1, const float* __restrict__ ad1,
    const float* __restrict__ as2, const float* __restrict__ ad2,
    float* __restrict__ es1, float* __restrict__ ed1,
    float* __restrict__ es2, float* __restrict__ ed2, int n)
{
    int i = blockIdx.x * blockDim.x + threadIdx.x;   // over n*HEADS*2
    int total = n * HEADS * 2;
    if (i >= total) return;
    int conv = (i >= n * HEADS) ? 1 : 0;
    int j = conv ? (i - n * HEADS) : i;
    int node = j >> 2;
    int h    = j & 3;

    const float* __restrict__ hp  = (conv ? H2 : H1) + node * F_OUT + h * C_OUT;
    const float* __restrict__ asp = (conv ? as2 : as1) + h * C_OUT;
    const float* __restrict__ adp = (conv ? ad2 : ad1) + h * C_OUT;

    float ssum = 0.0f, dsum = 0.0f;
    #pragma unroll
    for (int c = 0; c < C_OUT; ++c) {
        float v = hp[c];
        ssum += v * asp[c];
        dsum += v * adp[c];
    }
    if (conv) { es2[node * HEADS + h] = ssum; ed2[node * HEADS + h] = dsum; }
    else      { es1[node * HEADS + h] = ssum; ed1[node * HEADS + h] = dsum; }
}

// ---------------------------------------------------------------------------
// 3) init output with mixed bias; init segment max/denominator buffers
// ---------------------------------------------------------------------------
__global__ void init_kernel(
    const float* __restrict__ b1, const float* __restrict__ b2,
    float* __restrict__ out,
    float* __restrict__ m1, float* __restrict__ d1,
    float* __restrict__ m2, float* __restrict__ d2, int n)
{
    int i = blockIdx.x * blockDim.x + threadIdx.x;   // over n*F_OUT
    if (i >= n * F_OUT) return;
    int f = i & (F_OUT - 1);
    out[i] = (1.0f - ALPHA_) * b1[f] + ALPHA_ * b2[f];
    if (i < n * HEADS) {
        float ninf = -__builtin_huge_valf();
        m1[i] = ninf; m2[i] = ninf;
        d1[i] = 0.0f; d2[i] = 0.0f;
    }
}

// ---------------------------------------------------------------------------
// 4/5/6) edge passes. Items 0..E-1 are graph edges, items E..E+n-1 self loops.
// ---------------------------------------------------------------------------
__global__ void edge_max_kernel(
    const int* __restrict__ srcA, const int* __restrict__ dstA, int E, int n,
    const float* __restrict__ es, const float* __restrict__ ed,
    float* __restrict__ m)
{
    int idx = blockIdx.x * blockDim.x + threadIdx.x;   // over (E+n)*HEADS
    if (idx >= (E + n) * HEADS) return;
    int item = idx >> 2, h = idx & 3;
    int s, d;
    if (item < E) { s = srcA[item]; d = dstA[item]; }
    else          { s = d = item - E; }
    float v = lrelu(es[s * HEADS + h] + ed[d * HEADS + h]);
    atomicMaxFloat(&m[d * HEADS + h], v);
}

__global__ void edge_sum_kernel(
    const int* __restrict__ srcA, const int* __restrict__ dstA, int E, int n,
    const float* __restrict__ es, const float* __restrict__ ed,
    const float* __restrict__ m, float* __restrict__ denom)
{
    int idx = blockIdx.x * blockDim.x + threadIdx.x;
    if (idx >= (E + n) * HEADS) return;
    int item = idx >> 2, h = idx & 3;
    int s, d;
    if (item < E) { s = srcA[item]; d = dstA[item]; }
    else          { s = d = item - E; }
    float v = lrelu(es[s * HEADS + h] + ed[d * HEADS + h]);
    float ex = __expf(v - m[d * HEADS + h]);
    atomicAdd(&denom[d * HEADS + h], ex);
}

__global__ void edge_msg_kernel(
    const int* __restrict__ srcA, const int* __restrict__ dstA, int E, int n,
    const float* __restrict__ es, const float* __restrict__ ed,
    const float* __restrict__ m, const float* __restrict__ denom,
    const float* __restrict__ Hm, float scale, float* __restrict__ out)
{
    int idx = blockIdx.x * blockDim.x + threadIdx.x;
    if (idx >= (E + n) * HEADS) return;
    int item = idx >> 2, h = idx & 3;
    int s, d;
    if (item < E) { s = srcA[item]; d = dstA[item]; }
    else          { s = d = item - E; }
    float v = lrelu(es[s * HEADS + h] + ed[d * HEADS + h]);
    float att = __expf(v - m[d * HEADS + h]) / denom[d * HEADS + h];
    float f = scale * att;

    const float* __restrict__ hp = Hm + s * F_OUT + h * C_OUT;
    float* __restrict__ op       = out + d * F_OUT + h * C_OUT;
    __builtin_prefetch(hp, 0, 1);   // gfx1250 global_prefetch_b8
    #pragma unroll
    for (int c = 0; c < C_OUT; ++c) {
        atomicAdd(&op[c], f * hp[c]);
    }
}

// ---------------------------------------------------------------------------
extern "C" void kernel_launch(void* const* d_in, const int* in_sizes, int n_in,
                              void* d_out, int out_size, void* d_ws, size_t ws_size,
                              hipStream_t stream) {
    const float* x    = (const float*)d_in[0];
    const int*   ei   = (const int*)d_in[1];     // [2, E] row-major
    const float* W1   = (const float*)d_in[2];
    const float* as1  = (const float*)d_in[3];
    const float* ad1  = (const float*)d_in[4];
    const float* b1   = (const float*)d_in[5];
    const float* W2   = (const float*)d_in[6];
    const float* as2  = (const float*)d_in[7];
    const float* ad2  = (const float*)d_in[8];
    const float* b2   = (const float*)d_in[9];
    float* out = (float*)d_out;

    const int n = in_sizes[0] / F_IN;            // 50000
    const int E = in_sizes[1] / 2;               // 1600000
    const int* src = ei;
    const int* dst = ei + E;

    // Workspace layout (floats)
    float* p   = (float*)d_ws;
    float* H1  = p;  p += (size_t)n * F_OUT;
    float* H2  = p;  p += (size_t)n * F_OUT;
    float* es1 = p;  p += (size_t)n * HEADS;
    float* ed1 = p;  p += (size_t)n * HEADS;
    float* es2 = p;  p += (size_t)n * HEADS;
    float* ed2 = p;  p += (size_t)n * HEADS;
    float* m1  = p;  p += (size_t)n * HEADS;
    float* dn1 = p;  p += (size_t)n * HEADS;
    float* m2  = p;  p += (size_t)n * HEADS;
    float* dn2 = p;  p += (size_t)n * HEADS;

    // 1) WMMA GEMMs: one wave per 16x16 tile, both convs
    {
        int mtiles = (n + 15) / 16;
        int blocks = 2 * mtiles * (F_OUT / 16);
        gemm_wmma_kernel<<<blocks, 32, 0, stream>>>(x, W1, W2, H1, H2, n);
    }
    // 2) attention logits
    {
        int total = n * HEADS * 2;
        scores_kernel<<<(total + 255) / 256, 256, 0, stream>>>(
            H1, H2, as1, ad1, as2, ad2, es1, ed1, es2, ed2, n);
    }
    // 3) init out (mixed bias), max=-inf, denom=0
    {
        int total = n * F_OUT;
        init_kernel<<<(total + 255) / 256, 256, 0, stream>>>(
            b1, b2, out, m1, dn1, m2, dn2, n);
    }
    const int eth = (E + n) * HEADS;
    const int eblk = (eth + 255) / 256;
    // 4) segment max
    edge_max_kernel<<<eblk, 256, 0, stream>>>(src, dst, E, n, es1, ed1, m1);
    edge_max_kernel<<<eblk, 256, 0, stream>>>(dst, src, E, n, es2, ed2, m2);
    // 5) segment sum of exp
    edge_sum_kernel<<<eblk, 256, 0, stream>>>(src, dst, E, n, es1, ed1, m1, dn1);
    edge_sum_kernel<<<eblk, 256, 0, stream>>>(dst, src, E, n, es2, ed2, m2, dn2);
    // 6) weighted scatter: (1-A)*conv1 + A*conv2 directly into out
    edge_msg_kernel<<<eblk, 256, 0, stream>>>(src, dst, E, n, es1, ed1, m1, dn1,
                                              H1, 1.0f - ALPHA_, out);
    edge_msg_kernel<<<eblk, 256, 0, stream>>>(dst, src, E, n, es2, ed2, m2, dn2,
                                              H2, ALPHA_, out);
}